// PSRoIAlign_12687333392409
// MI455X (gfx1250) — compile-verified
//
#include <hip/hip_runtime.h>

// PSRoIAlign for MI455X (gfx1250, wave32).
// features: [B=2, C=490, H=160, W=160] f32; rois: [N,5] f32; stride: int scalar.
// out: [N, D=10, g=7, g=7] f32.
//
// Gather/latency-bound kernel. Block = (g, g, D) = (7,7,10) = 490 threads,
// one block per RoI: index decomposition is free (threadIdx), RoI params are
// block-uniform (scalarizable to SALU), and the 16 clamped corner gathers per
// output element are issued unconditionally as one batch so they resolve in
// parallel out of the 192MB L2 (the whole 100MB feature tensor is L2-resident).

#define G      7
#define SR     2
#define D_OUT  10
#define C_IN   490
#define H_IN   160
#define W_IN   160

__global__ __launch_bounds__(G * G * D_OUT) void psroi_align_gfx1250(
    const float* __restrict__ rois,
    const float* __restrict__ features,
    const int*   __restrict__ stride_p,
    float*       __restrict__ out)
{
    const int pw = threadIdx.x;   // 0..6  (fastest -> coalesced stores)
    const int ph = threadIdx.y;   // 0..6
    const int d  = threadIdx.z;   // 0..9
    const int n  = blockIdx.x;    // RoI id (block-uniform)

    const float scale = 1.0f / (float)stride_p[0];

    // ---- RoI parameters: uniform across the block -> SALU/const-cache ----
    const float* roi = rois + (size_t)n * 5;
    const int   b  = (int)roi[0];
    const float sw = roi[1] * scale - 0.5f;
    const float sh = roi[2] * scale - 0.5f;
    const float ew = roi[3] * scale - 0.5f;
    const float eh = roi[4] * scale - 0.5f;
    const float roi_w = fmaxf(ew - sw, 0.1f);
    const float roi_h = fmaxf(eh - sh, 0.1f);
    const float bin_w = roi_w * (1.0f / (float)G);
    const float bin_h = roi_h * (1.0f / (float)G);

    // position-sensitive channel: c = (d*g + ph)*g + pw
    const int c = (d * G + ph) * G + pw;
    const float* base = features + ((size_t)b * C_IN + (size_t)c) * (H_IN * W_IN);

    // ---- per-axis bilinear indices / fractions / validity ----
    int   ylo[SR], yhi[SR], xlo[SR], xhi[SR];
    float ly[SR], hy[SR], lx[SR], hx[SR];
    float my[SR], mx[SR];   // validity masks as 0/1 floats

    #pragma unroll
    for (int s = 0; s < SR; ++s) {
        const float off = ((float)s + 0.5f) * (1.0f / (float)SR);

        float y = sh + ((float)ph + off) * bin_h;
        my[s] = ((y > -1.0f) && (y < (float)H_IN)) ? 1.0f : 0.0f;
        float yc = fmaxf(y, 0.0f);
        int lo = (int)floorf(yc);
        lo = min(lo, H_IN - 1);
        ylo[s] = lo;
        yhi[s] = min(lo + 1, H_IN - 1);
        ly[s]  = (lo >= H_IN - 1) ? 0.0f : (yc - (float)lo);
        hy[s]  = 1.0f - ly[s];

        float x = sw + ((float)pw + off) * bin_w;
        mx[s] = ((x > -1.0f) && (x < (float)W_IN)) ? 1.0f : 0.0f;
        float xc = fmaxf(x, 0.0f);
        int lox = (int)floorf(xc);
        lox = min(lox, W_IN - 1);
        xlo[s] = lox;
        xhi[s] = min(lox + 1, W_IN - 1);
        lx[s]  = (lox >= W_IN - 1) ? 0.0f : (xc - (float)lox);
        hx[s]  = 1.0f - lx[s];
    }

    // ---- batch all 16 gathers (unconditional; addresses always clamped) ----
    float v00[SR][SR], v01[SR][SR], v10[SR][SR], v11[SR][SR];
    #pragma unroll
    for (int iy = 0; iy < SR; ++iy) {
        const float* r0 = base + ylo[iy] * W_IN;
        const float* r1 = base + yhi[iy] * W_IN;
        #pragma unroll
        for (int ix = 0; ix < SR; ++ix) {
            v00[iy][ix] = r0[xlo[ix]];
            v01[iy][ix] = r0[xhi[ix]];
            v10[iy][ix] = r1[xlo[ix]];
            v11[iy][ix] = r1[xhi[ix]];
        }
    }

    // ---- masked bilinear accumulate, mean over sr*sr ----
    float acc = 0.0f;
    #pragma unroll
    for (int iy = 0; iy < SR; ++iy) {
        #pragma unroll
        for (int ix = 0; ix < SR; ++ix) {
            float m = my[iy] * mx[ix];
            float v = hy[iy] * (hx[ix] * v00[iy][ix] + lx[ix] * v01[iy][ix])
                    + ly[iy] * (hx[ix] * v10[iy][ix] + lx[ix] * v11[iy][ix]);
            acc += m * v;
        }
    }

    // out[((n*D + d)*G + ph)*G + pw]
    out[((size_t)n * D_OUT + d) * (G * G) + ph * G + pw] =
        acc * (1.0f / (float)(SR * SR));
}

extern "C" void kernel_launch(void* const* d_in, const int* in_sizes, int n_in,
                              void* d_out, int out_size, void* d_ws, size_t ws_size,
                              hipStream_t stream) {
    const float* rois     = (const float*)d_in[0];
    const float* features = (const float*)d_in[1];
    const int*   stride_p = (const int*)d_in[2];
    float*       out      = (float*)d_out;

    const int N = in_sizes[0] / 5;               // 512 RoIs
    dim3 block(G, G, D_OUT);                     // 490 threads = 16 wave32s
    psroi_align_gfx1250<<<N, block, 0, stream>>>(rois, features, stride_p, out);
}